// FreqGrid_85839216377941
// MI455X (gfx1250) — compile-verified
//
#include <hip/hip_runtime.h>
#include <math.h>

// FreqGrid: factorized-plane feature sampling + cosine-basis contraction.
//
// MI455X reasoning:
//  * 100MB grid fits in 192MB L2 -> gather-throughput bound, not HBM bound.
//  * No shared contraction dim -> WMMA not applicable; win comes from halving
//    gather instruction count (align-4 b64 tap pairs), 32-bit GVS addressing
//    (SGPR base + VGPR offset), wave32 occupancy, and staging the 128-entry
//    basis table into LDS with the CDNA5 async global->LDS path.

#define RR   256
#define CC   16
#define FF   8
#define CHN  128   // C*F

typedef __attribute__((address_space(1))) int* gint_p;
typedef __attribute__((address_space(3))) int* lint_p;

// reflect(x, 256): span=255, m=510
__device__ __forceinline__ float reflectf(float x) {
    float r = fabsf(x);
    r = fmodf(r, 510.0f);
    return r > 255.0f ? (510.0f - r) : r;
}

// natural align(4) float pair -> eligible for global_load_b64 at align 4
struct f2 { float x, y; };

__device__ __forceinline__ void wait_asynccnt0() {
#if __has_builtin(__builtin_amdgcn_s_wait_asynccnt)
    __builtin_amdgcn_s_wait_asynccnt(0);
#else
    asm volatile("s_wait_asynccnt 0" ::: "memory");
#endif
}

__device__ __forceinline__ f2 ldp(const float* __restrict__ base, unsigned idx) {
    return *(const f2*)(base + idx);
}

__global__ __launch_bounds__(256)
void freqgrid_kernel(const float* __restrict__ coords,
                     const float* __restrict__ grid,
                     const float* __restrict__ freqs,
                     float* __restrict__ out,
                     int n_pts)
{
    __shared__ float s_basis[CHN];
    const int tid = threadIdx.x;

    // ---- Stage basis coefficients into LDS (CDNA5 async global->LDS) ----
    if (tid < CHN) {
        float fq;
#if __has_builtin(__builtin_amdgcn_global_load_async_to_lds_b32)
        __builtin_amdgcn_global_load_async_to_lds_b32(
            (gint_p)(uintptr_t)(freqs + tid),
            (lint_p)(uintptr_t)(&s_basis[tid]),
            0, 0);
        wait_asynccnt0();
        fq = s_basis[tid];
#else
        {
            unsigned lds_off = (unsigned)(uintptr_t)(&s_basis[tid]); // low 32b = LDS offset
            const float* gsrc = freqs + tid;
            asm volatile("global_load_async_to_lds_b32 %0, %1, off"
                         :: "v"(lds_off), "v"(gsrc) : "memory");
            wait_asynccnt0();
            fq = s_basis[tid];
        }
#endif
        fq = fminf(fmaxf(fq, 0.0f), 1.0f);
        float fr = exp2f(8.0f * fq) - 1.0f;                 // 2^(f*log2(256)) - 1
        s_basis[tid] = (3.14159265358979323846f / 256.0f) * (fr + 0.5f);
    }
    __syncthreads();

    const int n = blockIdx.x * blockDim.x + tid;
    if (n >= n_pts) return;

    const float c0 = coords[3 * n + 0];
    const float c1 = coords[3 * n + 1];
    const float c2 = coords[3 * n + 2];

    // pts_p + 0.5 (basis argument partner)
    const float q0 = (c0 + 1.0f) * 127.5f + 0.5f;
    const float q1 = (c1 + 1.0f) * 127.5f + 0.5f;
    const float q2 = (c2 + 1.0f) * 127.5f + 0.5f;

    // plane p samples (u->x, v->y): p0:(c1,c2)  p1:(c0,c2)  p2:(c0,c1)
    const float us[3] = {c1, c0, c0};
    const float vs[3] = {c2, c2, c1};

    unsigned off[3];
    float wx0[3], wx1[3], wy0[3], wy1[3];
#pragma unroll
    for (int p = 0; p < 3; ++p) {
        float ix = reflectf((us[p] + 1.0f) * 127.5f);   // in [0,255]
        float iy = reflectf((vs[p] + 1.0f) * 127.5f);
        int x0 = (int)ix; if (x0 > 254) x0 = 254;       // fold clamp: x1 = x0+1 always
        int y0 = (int)iy; if (y0 > 254) y0 = 254;
        float wx = ix - (float)x0;                      // wx==1 reproduces the x0=255 edge
        float wy = iy - (float)y0;
        wx1[p] = wx; wx0[p] = 1.0f - wx;
        wy1[p] = wy; wy0[p] = 1.0f - wy;
        off[p] = (unsigned)(y0 * RR + x0);
    }

    // Wave-uniform plane base pointers (SGPRs); per-lane index stays 32-bit
    // (max ~8.4M elements) -> GVS addressing: saddr + 32-bit voffset.
    const float* __restrict__ gp0 = grid;
    const float* __restrict__ gp1 = grid + (size_t)1 * CHN * RR * RR;
    const float* __restrict__ gp2 = grid + (size_t)2 * CHN * RR * RR;

    float* op = out + (size_t)n * CC;

#pragma unroll 1
    for (int c = 0; c < CC; ++c) {
        float acc = 0.0f;
#pragma unroll
        for (int f = 0; f < FF; ++f) {
            const int      ch = c * FF + f;
            const unsigned cs = (unsigned)ch << 16;     // ch * 256*256

            const float sb = s_basis[ch];

            // 2 rows x 2 contiguous taps per plane -> b64 gathers
            f2 r00 = ldp(gp0, cs + off[0]);
            f2 r01 = ldp(gp0, cs + off[0] + RR);
            f2 r10 = ldp(gp1, cs + off[1]);
            f2 r11 = ldp(gp1, cs + off[1] + RR);
            f2 r20 = ldp(gp2, cs + off[2]);
            f2 r21 = ldp(gp2, cs + off[2] + RR);

            float v0 = (r00.x * wx0[0] + r00.y * wx1[0]) * wy0[0]
                     + (r01.x * wx0[0] + r01.y * wx1[0]) * wy1[0];
            float v1 = (r10.x * wx0[1] + r10.y * wx1[1]) * wy0[1]
                     + (r11.x * wx0[1] + r11.y * wx1[1]) * wy1[1];
            float v2 = (r20.x * wx0[2] + r20.y * wx1[2]) * wy0[2]
                     + (r21.x * wx0[2] + r21.y * wx1[2]) * wy1[2];

            acc += v0 * __cosf(sb * q0) + v1 * __cosf(sb * q1) + v2 * __cosf(sb * q2);
        }
        op[c] = 2.0f * acc;
    }
}

extern "C" void kernel_launch(void* const* d_in, const int* in_sizes, int n_in,
                              void* d_out, int out_size, void* d_ws, size_t ws_size,
                              hipStream_t stream) {
    const float* coords = (const float*)d_in[0];
    const float* grid   = (const float*)d_in[1];
    const float* freqs  = (const float*)d_in[2];
    float* out = (float*)d_out;

    const int n = in_sizes[0] / 3;
    dim3 block(256);
    dim3 gridDim((n + 255) / 256);
    freqgrid_kernel<<<gridDim, block, 0, stream>>>(coords, grid, freqs, out, n);
}